// Memory_52974126628960
// MI455X (gfx1250) — compile-verified
//
#include <hip/hip_runtime.h>
#include <math.h>

// Problem constants (from reference): N = 1<<20 rows, W = 64.
#define WDIM 64

typedef __attribute__((ext_vector_type(2))) float v2f;
typedef __attribute__((ext_vector_type(8))) float v8f;

// ---------------------------------------------------------------------------
// Kernel 1: sim[row] = write_strength * (memory[row] . key) / (||row|| * ||key||)
// One wave handles a 16x64 tile via 16 x V_WMMA_F32_16X16X4_F32.
// A layout (32-bit A 16x4): lane L<16 -> row L, K={0,1}; lane L+16 -> row L, K={2,3}.
// B: only column N=0 (lanes with mrow==0) carries the key; selected branchlessly.
// ---------------------------------------------------------------------------
__global__ void sim_wmma_kernel(const float* __restrict__ mem,
                                const float* __restrict__ key,
                                const float* __restrict__ wstr,
                                float* __restrict__ sim) {
  __shared__ float s_dot[8][16];             // 8 waves per block
  const int tid  = threadIdx.x;
  const int wave = tid >> 5;
  const int lane = tid & 31;
  const int half = lane >> 4;                // 0: K={0,1}, 1: K={2,3}
  const int mrow = lane & 15;
  const long rowBase = ((long)blockIdx.x * 8 + wave) * 16;
  const long row = rowBase + mrow;
  const float* rp = mem + row * WDIM + half * 2;

  // Preload all 16 A fragments: 16 outstanding global_load_b64, no per-WMMA stalls.
  v2f afrag[16];
#pragma unroll
  for (int kk = 0; kk < 16; ++kk)
    afrag[kk] = *(const v2f*)(rp + kk * 4);

  // Preload B fragments branchlessly (v_cndmask, no exec masking / s_loads in loop).
  const bool col0 = (mrow == 0);
  v2f bfrag[16];
#pragma unroll
  for (int kk = 0; kk < 16; ++kk) {
    const float* kp = key + kk * 4 + half * 2;
    const float kx = kp[0];
    const float ky = kp[1];
    bfrag[kk].x = col0 ? kx : 0.f;
    bfrag[kk].y = col0 ? ky : 0.f;
  }

  // ||key|| (tiny, redundantly computed, served from scalar cache)
  float ksq = 0.f;
#pragma unroll
  for (int k = 0; k < WDIM; ++k) { float kv = key[k]; ksq += kv * kv; }
  const float kn = fmaxf(sqrtf(ksq), 1e-8f);

  // Back-to-back WMMA chain; row sum-of-squares co-executes on the VALU.
  v8f acc = {};
  float ss = 0.f;
#pragma unroll
  for (int kk = 0; kk < 16; ++kk) {
    ss += afrag[kk].x * afrag[kk].x + afrag[kk].y * afrag[kk].y;
    acc = __builtin_amdgcn_wmma_f32_16x16x4_f32(
        /*neg_a=*/false, afrag[kk], /*neg_b=*/false, bfrag[kk],
        /*c_mod=*/(short)0, acc, /*reuse_a=*/false, /*reuse_b=*/false);
  }

  // lane L and lane L^16 hold the two halves of row L's sum of squares
  ss += __shfl_xor(ss, 16, 32);
  const float mn = fmaxf(sqrtf(ss), 1e-8f);

  // D column N=0: lane 0 has rows 0..7 in acc[0..7], lane 16 has rows 8..15
  if (lane == 0) {
#pragma unroll
    for (int m = 0; m < 8; ++m) s_dot[wave][m] = acc[m];
  }
  if (lane == 16) {
#pragma unroll
    for (int m = 0; m < 8; ++m) s_dot[wave][8 + m] = acc[m];
  }
  __syncthreads();
  if (half == 0) {
    const float d = s_dot[wave][mrow];
    sim[row] = wstr[0] * d / (mn * kn);
  }
}

// ---------------------------------------------------------------------------
// Softmax: deterministic two-level reductions (no float atomics).
// ---------------------------------------------------------------------------
__global__ void reduce_max_kernel(const float* __restrict__ x, float* __restrict__ partial, int n) {
  __shared__ float sm[256];
  float m = -INFINITY;
  for (long i = (long)blockIdx.x * 256 + threadIdx.x; i < n; i += (long)gridDim.x * 256)
    m = fmaxf(m, x[i]);
  sm[threadIdx.x] = m; __syncthreads();
  for (int s = 128; s > 0; s >>= 1) {
    if (threadIdx.x < s) sm[threadIdx.x] = fmaxf(sm[threadIdx.x], sm[threadIdx.x + s]);
    __syncthreads();
  }
  if (threadIdx.x == 0) partial[blockIdx.x] = sm[0];
}

__global__ void final_max_kernel(const float* __restrict__ partial, float* __restrict__ gmax, int nb) {
  __shared__ float sm[256];
  float m = -INFINITY;
  for (int i = threadIdx.x; i < nb; i += 256) m = fmaxf(m, partial[i]);
  sm[threadIdx.x] = m; __syncthreads();
  for (int s = 128; s > 0; s >>= 1) {
    if (threadIdx.x < s) sm[threadIdx.x] = fmaxf(sm[threadIdx.x], sm[threadIdx.x + s]);
    __syncthreads();
  }
  if (threadIdx.x == 0) gmax[0] = sm[0];
}

__global__ void exp_sum_kernel(float* __restrict__ sim, const float* __restrict__ gmax,
                               float* __restrict__ partial, int n) {
  __shared__ float sm[256];
  const float gm = gmax[0];
  float s = 0.f;
  for (long i = (long)blockIdx.x * 256 + threadIdx.x; i < n; i += (long)gridDim.x * 256) {
    const float e = expf(sim[i] - gm);
    sim[i] = e;
    s += e;
  }
  sm[threadIdx.x] = s; __syncthreads();
  for (int st = 128; st > 0; st >>= 1) {
    if (threadIdx.x < st) sm[threadIdx.x] += sm[threadIdx.x + st];
    __syncthreads();
  }
  if (threadIdx.x == 0) partial[blockIdx.x] = sm[0];
}

__global__ void final_sum_kernel(const float* __restrict__ partial, float* __restrict__ gsum, int nb) {
  __shared__ float sm[256];
  float s = 0.f;
  for (int i = threadIdx.x; i < nb; i += 256) s += partial[i];
  sm[threadIdx.x] = s; __syncthreads();
  for (int st = 128; st > 0; st >>= 1) {
    if (threadIdx.x < st) sm[threadIdx.x] += sm[threadIdx.x + st];
    __syncthreads();
  }
  if (threadIdx.x == 0) gsum[0] = sm[0];
}

__global__ void normalize_kernel(const float* __restrict__ sim, const float* __restrict__ gsum,
                                 float* __restrict__ out, int n) {
  const long i = (long)blockIdx.x * 256 + threadIdx.x;
  if (i < n) out[i] = sim[i] / gsum[0];
}

// ---------------------------------------------------------------------------
// Allocation: stable LSD radix sort (float bits are monotonic for u >= 0),
// then exclusive prefix product + scatter.
// ---------------------------------------------------------------------------
__global__ void pack_kernel(const float* __restrict__ usage, unsigned* __restrict__ keys,
                            unsigned* __restrict__ idx, int n) {
  const long i = (long)blockIdx.x * 256 + threadIdx.x;
  if (i < n) { keys[i] = __float_as_uint(usage[i]); idx[i] = (unsigned)i; }
}

// Tile = 4096 elements per block, 256 threads, 16 elements/thread.
__global__ void radix_hist_kernel(const unsigned* __restrict__ keys, unsigned* __restrict__ hist,
                                  int nb, int shift) {
  __shared__ unsigned sh[256];
  const int t = threadIdx.x;
  sh[t] = 0u; __syncthreads();
  const long blk = blockIdx.x;
  for (int s = 0; s < 16; ++s) {
    const unsigned k = keys[blk * 4096 + s * 256 + t];
    atomicAdd(&sh[(k >> shift) & 255u], 1u);
  }
  __syncthreads();
  hist[(long)t * nb + blk] = sh[t];          // digit-major layout
}

__global__ void radix_scan_kernel(unsigned* __restrict__ hist, int nb) {
  __shared__ unsigned tot[256];
  __shared__ unsigned base[256];
  const int d = threadIdx.x;
  unsigned run = 0;
  for (int b = 0; b < nb; ++b) {
    const unsigned t = hist[(long)d * nb + b];
    hist[(long)d * nb + b] = run;
    run += t;
  }
  tot[d] = run;
  __syncthreads();
  if (d == 0) {
    unsigned acc = 0;
    for (int k = 0; k < 256; ++k) { base[k] = acc; acc += tot[k]; }
  }
  __syncthreads();
  const unsigned bd = base[d];
  for (int b = 0; b < nb; ++b) hist[(long)d * nb + b] += bd;
}

__global__ void radix_scatter_kernel(const unsigned* __restrict__ keysIn,
                                     const unsigned* __restrict__ idxIn,
                                     unsigned* __restrict__ keysOut,
                                     unsigned* __restrict__ idxOut,
                                     const unsigned* __restrict__ hist,
                                     int nb, int shift) {
  __shared__ unsigned cnt[256];
  __shared__ unsigned sdig[256];
  const int t = threadIdx.x;
  cnt[t] = 0u; __syncthreads();
  const long blk = blockIdx.x;
  for (int s = 0; s < 16; ++s) {
    const long i = blk * 4096 + s * 256 + t;
    const unsigned k  = keysIn[i];
    const unsigned id = idxIn[i];
    const unsigned d  = (k >> shift) & 255u;
    sdig[t] = d;
    __syncthreads();
    unsigned before = 0;
    for (int j = 0; j < t; ++j) before += (sdig[j] == d) ? 1u : 0u;
    const unsigned dst = hist[(long)d * nb + blk] + cnt[d] + before;
    keysOut[dst] = k;
    idxOut[dst]  = id;
    __syncthreads();
    atomicAdd(&cnt[d], 1u);
    __syncthreads();
  }
}

// Prefix product: tiles of 2048 (256 threads x 8 elems).
__global__ void tile_scan_kernel(const unsigned* __restrict__ keys,
                                 float* __restrict__ excl,
                                 float* __restrict__ blockProd) {
  __shared__ float sp[256];
  const int t = threadIdx.x;
  const long base = (long)blockIdx.x * 2048 + (long)t * 8;
  float lp[8];
  float p = 1.f;
#pragma unroll
  for (int k = 0; k < 8; ++k) { p *= __uint_as_float(keys[base + k]); lp[k] = p; }
  sp[t] = p; __syncthreads();
  // inclusive product scan over thread totals
  for (int off = 1; off < 256; off <<= 1) {
    const float x = (t >= off) ? sp[t - off] : 1.f;
    __syncthreads();
    sp[t] *= x;
    __syncthreads();
  }
  const float exclT = (t == 0) ? 1.f : sp[t - 1];
#pragma unroll
  for (int k = 0; k < 8; ++k)
    excl[base + k] = exclT * (k == 0 ? 1.f : lp[k - 1]);
  if (t == 255) blockProd[blockIdx.x] = sp[255];
}

__global__ void block_scan_kernel(const float* __restrict__ blockProd,
                                  float* __restrict__ blockBase, int nb) {
  if (threadIdx.x == 0) {
    float run = 1.f;
    for (int b = 0; b < nb; ++b) { blockBase[b] = run; run *= blockProd[b]; }
  }
}

__global__ void alloc_final_kernel(const unsigned* __restrict__ keys,
                                   const unsigned* __restrict__ idx,
                                   const float* __restrict__ excl,
                                   const float* __restrict__ blockBase,
                                   float* __restrict__ outAlloc, int n) {
  const long i = (long)blockIdx.x * 256 + threadIdx.x;
  if (i < n) {
    const float u = __uint_as_float(keys[i]);
    const float e = blockBase[i / 2048] * excl[i];
    outAlloc[idx[i]] = (1.f - u) * e;
  }
}

// ---------------------------------------------------------------------------
extern "C" void kernel_launch(void* const* d_in, const int* in_sizes, int n_in,
                              void* d_out, int out_size, void* d_ws, size_t ws_size,
                              hipStream_t stream) {
  const float* mem  = (const float*)d_in[0];   // (N, 64)
  const float* usg  = (const float*)d_in[1];   // (N, 1)
  const float* key  = (const float*)d_in[2];   // (1, 64)
  const float* wstr = (const float*)d_in[3];   // (1,)
  const int n = in_sizes[1];                   // N = 1<<20

  float* outAddr  = (float*)d_out;             // first N floats
  float* outAlloc = (float*)d_out + n;         // next N floats

  // Workspace carve-up (all 4B aligned)
  char* ws = (char*)d_ws;
  size_t off = 0;
  float*    sim   = (float*)   (ws + off); off += (size_t)n * 4;
  unsigned* keysA = (unsigned*)(ws + off); off += (size_t)n * 4;
  unsigned* idxA  = (unsigned*)(ws + off); off += (size_t)n * 4;
  unsigned* keysB = (unsigned*)(ws + off); off += (size_t)n * 4;
  unsigned* idxB  = (unsigned*)(ws + off); off += (size_t)n * 4;
  float*    excl  = (float*)   (ws + off); off += (size_t)n * 4;
  const int nbSort = n / 4096;                 // 256 radix blocks
  unsigned* hist  = (unsigned*)(ws + off); off += (size_t)256 * nbSort * 4;
  float* partialMax = (float*)(ws + off); off += 1024 * 4;
  float* partialSum = (float*)(ws + off); off += 1024 * 4;
  float* gmax = (float*)(ws + off); off += 4;
  float* gsum = (float*)(ws + off); off += 4;
  const int nbScan = n / 2048;                 // 512 scan tiles
  float* blockProd = (float*)(ws + off); off += (size_t)nbScan * 4;
  float* blockBase = (float*)(ws + off); off += (size_t)nbScan * 4;

  // --- cosine similarity via WMMA (16 rows per wave, 8 waves per block) ---
  sim_wmma_kernel<<<n / 128, 256, 0, stream>>>(mem, key, wstr, sim);

  // --- softmax ---
  reduce_max_kernel<<<1024, 256, 0, stream>>>(sim, partialMax, n);
  final_max_kernel<<<1, 256, 0, stream>>>(partialMax, gmax, 1024);
  exp_sum_kernel<<<1024, 256, 0, stream>>>(sim, gmax, partialSum, n);
  final_sum_kernel<<<1, 256, 0, stream>>>(partialSum, gsum, 1024);
  normalize_kernel<<<(n + 255) / 256, 256, 0, stream>>>(sim, gsum, outAddr, n);

  // --- allocation: radix sort on float bits ---
  pack_kernel<<<(n + 255) / 256, 256, 0, stream>>>(usg, keysA, idxA, n);
  unsigned* kin = keysA; unsigned* iin = idxA;
  unsigned* kout = keysB; unsigned* iout = idxB;
  for (int pass = 0; pass < 4; ++pass) {
    const int shift = pass * 8;
    radix_hist_kernel<<<nbSort, 256, 0, stream>>>(kin, hist, nbSort, shift);
    radix_scan_kernel<<<1, 256, 0, stream>>>(hist, nbSort);
    radix_scatter_kernel<<<nbSort, 256, 0, stream>>>(kin, iin, kout, iout, hist, nbSort, shift);
    unsigned* tk = kin; kin = kout; kout = tk;
    unsigned* ti = iin; iin = iout; iout = ti;
  }
  // after 4 passes, sorted data is back in keysA/idxA (== kin/iin)

  // --- exclusive prefix product over sorted usage, then scatter ---
  tile_scan_kernel<<<nbScan, 256, 0, stream>>>(kin, excl, blockProd);
  block_scan_kernel<<<1, 256, 0, stream>>>(blockProd, blockBase, nbScan);
  alloc_final_kernel<<<(n + 255) / 256, 256, 0, stream>>>(kin, iin, excl, blockBase, outAlloc, n);
}